// OrderingGenerator_9028021256848
// MI455X (gfx1250) — compile-verified
//
#include <hip/hip_runtime.h>
#include <hip/hip_bf16.h>

typedef __attribute__((ext_vector_type(2))) float v2f;
typedef __attribute__((ext_vector_type(8))) float v8f;
typedef __attribute__((ext_vector_type(4))) unsigned int u32x4;
typedef __attribute__((ext_vector_type(8))) int i32x8;
typedef __attribute__((ext_vector_type(4))) int i32x4;

#define H     256
#define NPROT 1024
#define NLIG  128
#define NRES  256
#define NMOT  32
#define PFD   27
#define LFD   16
#define NB    64

// workspace layout (floats)
#define WS_VPROT  0      // 32 (27 + 5 zero pad)
#define WS_VLIG   32     // 16
#define WS_VTOK   48     // 16
#define WS_ERES   64     // 21
#define WS_EMOT   96     // 501
#define WS_BIAS   600    // 3 scalars: per-unit bias dots
#define WS_PROTS  640    // 64
#define WS_LIGS   704    // 64
#define WS_RESS   768    // 64
#define WS_TOKS   832    // 2048

// ---------------- Kernel A: tiny projection GEMVs + zero accumulators ----------------
__global__ void precompute_kernel(const float* __restrict__ Wp, const float* __restrict__ Wl,
                                  const float* __restrict__ emb, const float* __restrict__ remb,
                                  const float* __restrict__ Wm, const float* __restrict__ bp,
                                  const float* __restrict__ bl, float* __restrict__ ws) {
  if (blockIdx.x == 0) {
    for (int i = threadIdx.x; i < 5; i += blockDim.x)    ws[WS_VPROT + 27 + i] = 0.f;
    for (int i = threadIdx.x; i < 64; i += blockDim.x)   ws[WS_PROTS + i] = 0.f;
    for (int i = threadIdx.x; i < 64; i += blockDim.x)   ws[WS_LIGS + i] = 0.f;
    for (int i = threadIdx.x; i < 2048; i += blockDim.x) ws[WS_TOKS + i] = 0.f;
  }
  const int lane = threadIdx.x & 31;
  const int wid  = blockIdx.x * (blockDim.x >> 5) + (threadIdx.x >> 5);
  const int nw   = gridDim.x * (blockDim.x >> 5);
  for (int r = wid; r < 584; r += nw) {
    const float* src; const float* w; float* dst;
    if (r < 27)        { src = Wp   + r * H;        w = Wm;        dst = ws + WS_VPROT + r; }
    else if (r < 43)   { src = Wl   + (r - 27) * H; w = Wm + 512;  dst = ws + WS_VLIG + (r - 27); }
    else if (r < 59)   { src = Wl   + (r - 43) * H; w = Wm + 1024; dst = ws + WS_VTOK + (r - 43); }
    else if (r < 80)   { src = remb + (r - 59) * H; w = Wm + 256;  dst = ws + WS_ERES + (r - 59); }
    else if (r < 581)  { src = emb  + (r - 80) * H; w = Wm + 768;  dst = ws + WS_EMOT + (r - 80); }
    else if (r == 581) { src = bp; w = Wm;        dst = ws + WS_BIAS + 0; }
    else if (r == 582) { src = bl; w = Wm + 512;  dst = ws + WS_BIAS + 1; }
    else               { src = bl; w = Wm + 1024; dst = ws + WS_BIAS + 2; }
    float acc = 0.f;
    for (int i = lane; i < H; i += 32) acc += src[i] * w[i];
    for (int o = 16; o; o >>= 1) acc += __shfl_down(acc, o, 32);
    if (lane == 0) *dst = acc;
  }
}

// ---------------- Kernel B: TDM tile -> LDS -> V_WMMA_F32_16X16X4_F32 ----------------
// Each block: one contiguous 128-atom x 27-float tile (13824 B) DMA'd by wave 0 via
// tensor_load_to_lds; all 8 waves then run ones-GEMM feature sums out of LDS.
__global__ void protein_kernel(const float* __restrict__ F, float* __restrict__ ws) {
  __shared__ float tile[128 * PFD];

  const int lane   = threadIdx.x & 31;
  const int wave   = threadIdx.x >> 5;
  const int sample = blockIdx.x >> 3;   // 8 blocks per sample
  const int split  = blockIdx.x & 7;

  if (threadIdx.x < 32) {
    // Build Tensor DMA Descriptor (D#) for a 2D tile: 27 elems/row x 128 rows, 4B elems.
    unsigned long long ga =
        (unsigned long long)(uintptr_t)(F + (size_t)(sample * NPROT + split * 128) * PFD);
    unsigned int lds_off = (unsigned int)(uintptr_t)(&tile[0]);   // low 32 bits = LDS offset
    u32x4 g0;
    g0[0] = 1u;                                                   // count=1, user mode
    g0[1] = lds_off;                                              // lds_addr [63:32]
    g0[2] = (unsigned int)(ga & 0xFFFFFFFFu);                     // global_addr low
    g0[3] = (unsigned int)((ga >> 32) & 0x1FFFFFFu) | (2u << 30); // addr hi + type=2
    i32x8 g1;
    g1[0] = 0x00020000;            // workgroup_mask=0, data_size=2 (4 bytes)
    g1[1] = (int)(27u << 16);      // tensor_dim0[15:0] in bits 63:48
    g1[2] = (int)(128u << 16);     // tensor_dim0 hi=0; tensor_dim1[15:0]=128
    g1[3] = (int)(27u << 16);      // tensor_dim1 hi=0; tile_dim0=27
    g1[4] = 128;                   // tile_dim1=128, tile_dim2=0
    g1[5] = 27;                    // tensor_dim0_stride low 32 = 27 elems
    g1[6] = 0;                     // stride hi, tensor_dim1_stride lo
    g1[7] = 0;
    i32x4 g2 = {0, 0, 0, 0};
    i32x4 g3 = {0, 0, 0, 0};
#if defined(__clang_major__) && (__clang_major__ >= 23)
    i32x8 g4 = {0, 0, 0, 0, 0, 0, 0, 0};
    __builtin_amdgcn_tensor_load_to_lds(g0, g1, g2, g3, g4, 0);
#else
    __builtin_amdgcn_tensor_load_to_lds(g0, g1, g2, g3, 0);
#endif
    __builtin_amdgcn_s_wait_tensorcnt(0);
  }
  __syncthreads();

  const int col   = lane & 15;
  const int colHi = col + 16;
  const int colC  = (colHi < PFD) ? colHi : (PFD - 1);  // clamped: vph==0 kills pad cols
  const int k0    = (lane < 16) ? 0 : 2;                // B VGPR0 holds K={0,2}, VGPR1 K={1,3}
  const float vpl = ws[WS_VPROT + col];
  const float vph = ws[WS_VPROT + colHi];               // zero for colHi >= 27

  v2f a = {1.f, 1.f};                 // all-ones A matrix (16x4)
  v8f acc0 = {};                      // feats 0..15
  v8f acc1 = {};                      // feats 16..31 (pad cols nulled by vph)

  for (int it = 0; it < 4; ++it) {    // 4 atoms per WMMA, 16 atoms per wave
    const int ab = (wave * 16 + it * 4 + k0) * PFD;
    v2f blo, bhi;
    blo[0] = tile[ab + col];
    blo[1] = tile[ab + PFD + col];
    bhi[0] = tile[ab + colC];
    bhi[1] = tile[ab + PFD + colC];
    acc0 = __builtin_amdgcn_wmma_f32_16x16x4_f32(false, a, false, blo, (short)0, acc0, false, false);
    acc1 = __builtin_amdgcn_wmma_f32_16x16x4_f32(false, a, false, bhi, (short)0, acc1, false, false);
  }
  // acc[0]: lanes 0-15 -> M=0 row (feature sums); lanes 16-31 -> M=8 row (identical, mask out)
  float t = acc0[0] * vpl + acc1[0] * vph;
  if (lane >= 16) t = 0.f;
  for (int o = 16; o; o >>= 1) t += __shfl_down(t, o, 32);
  if (lane == 0) atomicAdd(&ws[WS_PROTS + sample], t);
}

// ---------------- Kernel C: ligand pooled dot + motif token scatter ----------------
__device__ __forceinline__ float dot16(const float* __restrict__ row, const float* __restrict__ v) {
  const float4* r = (const float4*)row;
  float4 f0 = r[0], f1 = r[1], f2 = r[2], f3 = r[3];
  return f0.x*v[0] + f0.y*v[1] + f0.z*v[2] + f0.w*v[3]
       + f1.x*v[4] + f1.y*v[5] + f1.z*v[6] + f1.w*v[7]
       + f2.x*v[8] + f2.y*v[9] + f2.z*v[10] + f2.w*v[11]
       + f3.x*v[12] + f3.y*v[13] + f3.z*v[14] + f3.w*v[15];
}

__global__ void ligand_kernel(const float* __restrict__ FL,
                              const int* __restrict__ batch_ligand,
                              const int* __restrict__ motif_atoms,
                              const int* __restrict__ motif_atom_index,
                              const int* __restrict__ motif_atoms_batch,
                              float* __restrict__ ws) {
  const int j = blockIdx.x * blockDim.x + threadIdx.x;
  if (j >= NB * NLIG) return;
  // pooled ligand representation contribution of atom j
  float dl = dot16(FL + (size_t)j * LFD, ws + WS_VLIG);
  atomicAdd(&ws[WS_LIGS + batch_ligand[j]], dl);
  // token (motif) contribution of motif-atom entry j
  const int s = motif_atoms_batch[j];
  const int g = s * NLIG + motif_atoms[j];
  float dt = dot16(FL + (size_t)g * LFD, ws + WS_VTOK);
  atomicAdd(&ws[WS_TOKS + s * NMOT + motif_atom_index[j]], dt);
}

// ---------------- Kernel D: residue table-lookup sum per sample ----------------
__global__ void residue_kernel(const int* __restrict__ aa, float* __restrict__ ws) {
  __shared__ float eres[32];
  __shared__ float partial[8];
  if (threadIdx.x < 21) eres[threadIdx.x] = ws[WS_ERES + threadIdx.x];
  __syncthreads();
  const int b = blockIdx.x;
  float v = eres[aa[b * NRES + threadIdx.x]];
  const int lane = threadIdx.x & 31;
  for (int o = 16; o; o >>= 1) v += __shfl_down(v, o, 32);
  if (lane == 0) partial[threadIdx.x >> 5] = v;
  __syncthreads();
  if (threadIdx.x == 0) {
    float s = 0.f;
    for (int i = 0; i < 8; ++i) s += partial[i];
    ws[WS_RESS + b] = s;
  }
}

// ---------------- Kernel E: logits + softmax + full descending sort (top_k) ----------
__global__ void finalize_kernel(const int* __restrict__ wids, const float* __restrict__ bm,
                                const float* __restrict__ ws, float* __restrict__ out) {
  const int b = blockIdx.x;
  const int m = threadIdx.x;   // 0..31, one wave per sample
  const float bias = 1024.f * ws[WS_BIAS + 0] + 128.f * ws[WS_BIAS + 1]
                   + 4.f * ws[WS_BIAS + 2] + bm[0];
  float logit = ws[WS_PROTS + b] + ws[WS_RESS + b] + ws[WS_LIGS + b]
              + ws[WS_EMOT + wids[b * NMOT + m]] + ws[WS_TOKS + b * NMOT + m] + bias;
  // softmax over 32 motifs
  float mx = logit;
  for (int o = 16; o; o >>= 1) mx = fmaxf(mx, __shfl_xor(mx, o, 32));
  float e = expf(logit - mx);
  float s = e;
  for (int o = 16; o; o >>= 1) s += __shfl_xor(s, o, 32);
  float v = e / s;
  int idx = m;
  // bitonic sort: descending values, stable ties -> ascending index (matches jax.lax.top_k)
  for (int k = 2; k <= 32; k <<= 1) {
    for (int jj = k >> 1; jj; jj >>= 1) {
      float ov = __shfl_xor(v, jj, 32);
      int   oi = __shfl_xor(idx, jj, 32);
      bool lower   = ((m & jj) == 0);
      bool descBlk = ((m & k) == 0);
      bool iBeats  = (v > ov) || (v == ov && idx < oi);
      bool keepMine = lower ? (descBlk ? iBeats : !iBeats)
                            : (descBlk ? !iBeats : iBeats);
      if (!keepMine) { v = ov; idx = oi; }
    }
  }
  out[b * NMOT + m] = v;                                // vals [64,32]
  out[NB * NMOT + b * NMOT + m] = (float)idx;           // idx  [64,32]
}

extern "C" void kernel_launch(void* const* d_in, const int* in_sizes, int n_in,
                              void* d_out, int out_size, void* d_ws, size_t ws_size,
                              hipStream_t stream) {
  const float* prot_feat = (const float*)d_in[1];
  const float* lig_feat  = (const float*)d_in[2];
  const int*   batch_lig = (const int*)d_in[4];
  const int*   amino     = (const int*)d_in[5];
  const int*   wids      = (const int*)d_in[7];
  const int*   m_atoms   = (const int*)d_in[9];
  const int*   m_aidx    = (const int*)d_in[10];
  const int*   m_batch   = (const int*)d_in[11];
  const float* W_prot    = (const float*)d_in[12];
  const float* b_prot    = (const float*)d_in[13];
  const float* W_lig     = (const float*)d_in[14];
  const float* b_lig     = (const float*)d_in[15];
  const float* emb       = (const float*)d_in[16];
  const float* remb      = (const float*)d_in[17];
  const float* W_mlp     = (const float*)d_in[18];
  const float* b_mlp     = (const float*)d_in[19];
  float* ws  = (float*)d_ws;
  float* out = (float*)d_out;

  precompute_kernel<<<32, 256, 0, stream>>>(W_prot, W_lig, emb, remb, W_mlp, b_prot, b_lig, ws);
  protein_kernel<<<NB * 8, 256, 0, stream>>>(prot_feat, ws);
  ligand_kernel<<<(NB * NLIG) / 256, 256, 0, stream>>>(lig_feat, batch_lig, m_atoms, m_aidx, m_batch, ws);
  residue_kernel<<<NB, 256, 0, stream>>>(amino, ws);
  finalize_kernel<<<NB, 32, 0, stream>>>(wids, b_mlp, ws, out);
}